// GNN_4355096838211
// MI455X (gfx1250) — compile-verified
//
#include <hip/hip_runtime.h>
#include <hip/hip_bf16.h>

// ---------------------------------------------------------------------------
// GCN forward for MI455X (gfx1250, wave32).
//   - fp32 WMMA (v_wmma_f32_16x16x4_f32) for the node-feature GEMMs (exact
//     precision; GEMMs are bandwidth-bound so low-precision buys nothing).
//   - GEMM epilogue fuses the self-loop term (agg = h*dinv^2), saving a full
//     51 MB read+write pass per layer.
//   - wave-per-edge gather/scatter with f32 global atomics (L2-resident).
//   - wave-per-node LayerNorm+ReLU with wave32 shfl reductions.
// ---------------------------------------------------------------------------

#define N_NODES  100000
#define N_EDGES  1600000
#define N_GRAPHS 512
#define FEAT     128
#define OUTF     64

typedef __attribute__((ext_vector_type(2))) float v2f;
typedef __attribute__((ext_vector_type(8))) float v8f;

// ---------------------------------------------------------------------------
// Tiled GEMM: out[M x N] = A[M x 128] * W[128 x N] (+ bias[N])
// N is compile-time so every inner-loop access is base + immediate offset
// (no 64-bit address math between WMMAs).
// Block = 256 threads = 8 waves; block covers 16 rows, wave w covers columns
// [16w, 16w+16). A-tile (16x128 fp32) staged in LDS, padded stride 132
// (bank stride 4 -> conflict-free). K loop = 32 x wmma 16x16x4.
// If aggInit != null, also writes aggInit = out * dinv[row]^2 (self-loop).
// ---------------------------------------------------------------------------
template <int N>
__global__ __launch_bounds__(256)
void gemm_wmma_f32(const float* __restrict__ A, const float* __restrict__ W,
                   const float* __restrict__ bias, float* __restrict__ out,
                   float* __restrict__ aggInit, const float* __restrict__ dinv) {
    __shared__ float lds[16 * 132];

    const int tid     = threadIdx.x;
    const int rowBase = blockIdx.x * 16;

    // Cooperative load of the 16x128 A tile: 2048 floats, 8 per thread.
    {
        int e = tid * 8;
        int r = e >> 7;          // 0..15
        int c = e & 127;         // multiple of 8
        const float4* src = (const float4*)(A + (rowBase + r) * FEAT + c);
        float4 p0 = src[0];
        float4 p1 = src[1];
        float* d = &lds[r * 132 + c];
        d[0] = p0.x; d[1] = p0.y; d[2] = p0.z; d[3] = p0.w;
        d[4] = p1.x; d[5] = p1.y; d[6] = p1.z; d[7] = p1.w;
    }
    __syncthreads();

    const int wave    = tid >> 5;
    const int lane    = tid & 31;
    const int colBase = wave * 16;
    if (colBase >= N) return;            // wave-uniform; EXEC stays all-ones

    const int r  = lane & 15;            // A row within tile / C column
    const int kh = lane >> 4;            // K half-select (0: K pair 0, 1: pair 1)
    const int c  = lane & 15;            // output column within tile

    v8f acc = {0.f, 0.f, 0.f, 0.f, 0.f, 0.f, 0.f, 0.f};

    // Loop-invariant bases; per-step offsets are compile-time immediates.
    const float* wp = W + kh * 2 * N + colBase + c;   // B fragment base
    const float* ap = &lds[r * 132 + kh * 2];         // A fragment base

#pragma unroll
    for (int kk = 0; kk < 32; ++kk) {
        v2f av;
        av[0] = ap[kk * 4];
        av[1] = ap[kk * 4 + 1];
        v2f bv;
        bv[0] = wp[kk * 4 * N];
        bv[1] = wp[kk * 4 * N + N];
        acc = __builtin_amdgcn_wmma_f32_16x16x4_f32(
            /*neg_a=*/false, av, /*neg_b=*/false, bv,
            /*c_mod=*/(short)0, acc, /*reuse_a=*/false, /*reuse_b=*/false);
    }

    const float bcol = bias ? bias[colBase + c] : 0.f;
    const int   orow = rowBase + kh * 8;              // C/D layout: VGPR v -> M=v(+8)
    float* op = out + orow * N + colBase + c;

    if (aggInit) {
        float* gp = aggInit + orow * N + colBase + c;
#pragma unroll
        for (int v = 0; v < 8; ++v) {
            float val = acc[v] + bcol;
            op[v * N] = val;
            float dv = dinv[orow + v];
            gp[v * N] = val * dv * dv;
        }
    } else {
#pragma unroll
        for (int v = 0; v < 8; ++v) {
            op[v * N] = acc[v] + bcol;
        }
    }
}

// ---------------------------------------------------------------------------
// Helpers
// ---------------------------------------------------------------------------
__global__ __launch_bounds__(256)
void fill_f32(float* __restrict__ p, float v, int n) {
    int i = blockIdx.x * blockDim.x + threadIdx.x;
    if (i < n) p[i] = v;
}

__global__ __launch_bounds__(256)
void deg_accum(const int* __restrict__ dst, float* __restrict__ deg, int nE) {
    int i = blockIdx.x * blockDim.x + threadIdx.x;
    if (i < nE) atomicAdd(&deg[dst[i]], 1.0f);
}

__global__ __launch_bounds__(256)
void rsqrt_inplace(float* __restrict__ p, int n) {
    int i = blockIdx.x * blockDim.x + threadIdx.x;
    if (i < n) p[i] = rsqrtf(p[i]);
}

// wave-per-edge: agg[dst] += h[src] * dinv[src]*dinv[dst]
__global__ __launch_bounds__(256)
void edge_scatter(const float* __restrict__ h, const int* __restrict__ src,
                  const int* __restrict__ dst, const float* __restrict__ dinv,
                  float* __restrict__ agg, int nE) {
    int gid  = blockIdx.x * blockDim.x + threadIdx.x;
    int e    = gid >> 5;
    int lane = threadIdx.x & 31;
    if (e >= nE) return;
    int s = src[e], d = dst[e];
    float coef = dinv[s] * dinv[d];
    float4 v = ((const float4*)(h + (size_t)s * FEAT))[lane];
    float* ap = agg + (size_t)d * FEAT + lane * 4;
    atomicAdd(ap + 0, v.x * coef);
    atomicAdd(ap + 1, v.y * coef);
    atomicAdd(ap + 2, v.z * coef);
    atomicAdd(ap + 3, v.w * coef);
}

// wave-per-node: row = relu(layernorm(row + b) * g + be), in place
__global__ __launch_bounds__(256)
void ln_relu(float* __restrict__ agg, const float* __restrict__ bias,
             const float* __restrict__ g, const float* __restrict__ be, int nN) {
    int gid  = blockIdx.x * blockDim.x + threadIdx.x;
    int n    = gid >> 5;
    int lane = threadIdx.x & 31;
    if (n >= nN) return;
    float4* row = (float4*)(agg + (size_t)n * FEAT);
    float4 v  = row[lane];
    float4 bb = ((const float4*)bias)[lane];
    v.x += bb.x; v.y += bb.y; v.z += bb.z; v.w += bb.w;

    float s = v.x + v.y + v.z + v.w;
#pragma unroll
    for (int m = 16; m >= 1; m >>= 1) s += __shfl_xor(s, m, 32);
    float mean = s * (1.0f / 128.0f);

    float dx = v.x - mean, dy = v.y - mean, dz = v.z - mean, dw = v.w - mean;
    float q = dx * dx + dy * dy + dz * dz + dw * dw;
#pragma unroll
    for (int m = 16; m >= 1; m >>= 1) q += __shfl_xor(q, m, 32);
    float inv = rsqrtf(q * (1.0f / 128.0f) + 1e-5f);

    float4 gg = ((const float4*)g)[lane];
    float4 ee = ((const float4*)be)[lane];
    float4 o;
    o.x = fmaxf(dx * inv * gg.x + ee.x, 0.f);
    o.y = fmaxf(dy * inv * gg.y + ee.y, 0.f);
    o.z = fmaxf(dz * inv * gg.z + ee.z, 0.f);
    o.w = fmaxf(dw * inv * gg.w + ee.w, 0.f);
    row[lane] = o;
}

// wave-per-node: pooled[batch[n]] += h[n]
__global__ __launch_bounds__(256)
void pool_add(const float* __restrict__ h, const int* __restrict__ batch,
              float* __restrict__ pooled, int nN) {
    int gid  = blockIdx.x * blockDim.x + threadIdx.x;
    int n    = gid >> 5;
    int lane = threadIdx.x & 31;
    if (n >= nN) return;
    int b = batch[n];
    float4 v = ((const float4*)(h + (size_t)n * FEAT))[lane];
    float* pp = pooled + (size_t)b * FEAT + lane * 4;
    atomicAdd(pp + 0, v.x);
    atomicAdd(pp + 1, v.y);
    atomicAdd(pp + 2, v.z);
    atomicAdd(pp + 3, v.w);
}

// ---------------------------------------------------------------------------
extern "C" void kernel_launch(void* const* d_in, const int* in_sizes, int n_in,
                              void* d_out, int out_size, void* d_ws, size_t ws_size,
                              hipStream_t stream) {
    (void)in_sizes; (void)n_in; (void)out_size; (void)ws_size;

    const float* x     = (const float*)d_in[0];
    const int*   ei    = (const int*)d_in[1];         // [2, E] flat
    const int*   batch = (const int*)d_in[2];
    const float* W0 = (const float*)d_in[3];
    const float* b0 = (const float*)d_in[4];
    const float* g0 = (const float*)d_in[5];
    const float* e0 = (const float*)d_in[6];
    const float* W1 = (const float*)d_in[7];
    const float* b1 = (const float*)d_in[8];
    const float* g1 = (const float*)d_in[9];
    const float* e1 = (const float*)d_in[10];
    const float* W2 = (const float*)d_in[11];
    const float* b2 = (const float*)d_in[12];
    const float* g2 = (const float*)d_in[13];
    const float* e2 = (const float*)d_in[14];
    const float* Wf = (const float*)d_in[15];
    const float* bf = (const float*)d_in[16];

    const int* src = ei;
    const int* dst = ei + N_EDGES;

    // Workspace layout (bytes): dinv | bufA | bufB | pooled
    char* ws = (char*)d_ws;
    float* dinv   = (float*)ws;                                  // N floats
    float* bufA   = (float*)(ws + 400384);                       // N*128 floats
    float* bufB   = bufA + (size_t)N_NODES * FEAT;               // N*128 floats
    float* pooled = bufB + (size_t)N_NODES * FEAT;               // 512*128 floats

    const int TPB = 256;
    const int nodeBlocks  = (N_NODES + TPB - 1) / TPB;
    const int edgeBlocks  = (N_EDGES + TPB - 1) / TPB;
    const int waveNodeBlk = (N_NODES * 32 + TPB - 1) / TPB;
    const int waveEdgeBlk = (N_EDGES / 8);                       // E*32/256 exact

    // Degrees: deg = 1 (self loop) + sum over dst; then dinv = rsqrt(deg)
    fill_f32<<<nodeBlocks, TPB, 0, stream>>>(dinv, 1.0f, N_NODES);
    deg_accum<<<edgeBlocks, TPB, 0, stream>>>(dst, dinv, N_EDGES);
    rsqrt_inplace<<<nodeBlocks, TPB, 0, stream>>>(dinv, N_NODES);

    const float* Ws[3]  = {W0, W1, W2};
    const float* bs[3]  = {b0, b1, b2};
    const float* gs[3]  = {g0, g1, g2};
    const float* bes[3] = {e0, e1, e2};

    const float* cur = x;                 // layer input (bufA after layer 0)
    for (int l = 0; l < 3; ++l) {
        // h = cur @ W -> bufB, fused agg init: bufA = h*dinv^2
        gemm_wmma_f32<FEAT><<<N_NODES / 16, TPB, 0, stream>>>(
            cur, Ws[l], nullptr, bufB, bufA, dinv);
        // agg[dst] += h[src]*dinv[src]*dinv[dst]
        edge_scatter<<<waveEdgeBlk, TPB, 0, stream>>>(bufB, src, dst, dinv, bufA, N_EDGES);
        // bufA = relu(LN(agg + b))
        ln_relu<<<waveNodeBlk, TPB, 0, stream>>>(bufA, bs[l], gs[l], bes[l], N_NODES);
        cur = bufA;
    }

    // Global add pool, then head GEMM (with bias) straight into d_out.
    fill_f32<<<(N_GRAPHS * FEAT + TPB - 1) / TPB, TPB, 0, stream>>>(pooled, 0.0f, N_GRAPHS * FEAT);
    pool_add<<<waveNodeBlk, TPB, 0, stream>>>(bufA, batch, pooled, N_NODES);
    gemm_wmma_f32<OUTF><<<N_GRAPHS / 16, TPB, 0, stream>>>(
        pooled, Wf, bf, (float*)d_out, nullptr, nullptr);
}